// ConvLogicNetCIFAR_81123342287627
// MI455X (gfx1250) — compile-verified
//
#include <hip/hip_runtime.h>

#define DEV_INLINE __device__ __forceinline__

// COEF table from the reference (16 gates x 4 coefficients)
__constant__ float COEF_TBL[64] = {
  0.f, 0.f, 0.f, 0.f,   0.f, 0.f, 0.f, 1.f,
  0.f, 1.f, 0.f,-1.f,   0.f, 1.f, 0.f, 0.f,
  0.f, 0.f, 1.f,-1.f,   0.f, 0.f, 1.f, 0.f,
  0.f, 1.f, 1.f,-2.f,   0.f, 1.f, 1.f,-1.f,
  1.f,-1.f,-1.f, 1.f,   1.f,-1.f,-1.f, 2.f,
  1.f, 0.f,-1.f, 0.f,   1.f, 0.f,-1.f, 1.f,
  1.f,-1.f, 0.f, 0.f,   1.f,-1.f, 0.f, 1.f,
  1.f, 0.f, 0.f,-1.f,   1.f, 0.f, 0.f, 0.f
};

// ---------------------------------------------------------------------------
// coef[g][0..3] = softmax(w[g][0..15]) @ COEF
// ---------------------------------------------------------------------------
__global__ void softmax_coef_kernel(const float* __restrict__ w,
                                    float* __restrict__ coef, int n_gates) {
  int g = blockIdx.x * blockDim.x + threadIdx.x;
  if (g >= n_gates) return;
  const float* wp = w + (size_t)g * 16;
  float wv[16];
  float mx = wp[0];
  #pragma unroll
  for (int t = 0; t < 16; ++t) { wv[t] = wp[t]; mx = fmaxf(mx, wv[t]); }
  float e[16], s = 0.f;
  #pragma unroll
  for (int t = 0; t < 16; ++t) { e[t] = expf(wv[t] - mx); s += e[t]; }
  float inv = 1.0f / s;
  float c0 = 0.f, c1 = 0.f, c2 = 0.f, c3 = 0.f;
  #pragma unroll
  for (int t = 0; t < 16; ++t) {
    float p = e[t] * inv;
    c0 += p * COEF_TBL[t*4+0];
    c1 += p * COEF_TBL[t*4+1];
    c2 += p * COEF_TBL[t*4+2];
    c3 += p * COEF_TBL[t*4+3];
  }
  coef[(size_t)g*4+0] = c0; coef[(size_t)g*4+1] = c1;
  coef[(size_t)g*4+2] = c2; coef[(size_t)g*4+3] = c3;
}

// ---------------------------------------------------------------------------
// Soft logic gate: c0 + c1*a + c2*b + c3*(a*b)
// ---------------------------------------------------------------------------
DEV_INLINE float gate4(float c0, float c1, float c2, float c3, float a, float b) {
  return c0 + c1 * a + c2 * b + c3 * (a * b);
}

// Shared conv-tree + orpool evaluation over an LDS-resident input plane.
// Input plane s: CIN x H x H floats for one image. Output: OC x (H/2) x (H/2).
// NOTE: reference uses coef rows 0..3 only (off = 2^level - 1 slicing):
//   level0 -> rows 0,1,2,3 ; level1 -> rows 1,2 ; level2 -> row 3
DEV_INLINE void conv_eval(const float* s, const int* __restrict__ leaf,
                          const float* __restrict__ coef, float* __restrict__ out,
                          int b, int H, int OC, int transposed) {
  const int Hp = H >> 1;
  const int PP = Hp * Hp;
  const int total = OC * PP;
  const int chunk = total / gridDim.y;
  const int o_beg = chunk * blockIdx.y;
  const int o_end = o_beg + chunk;
  for (int o = o_beg + threadIdx.x; o < o_end; o += blockDim.x) {
    const int c  = o / PP;
    const int pp = o - c * PP;
    const int pi = pp / Hp;
    const int pj = pp - pi * Hp;
    int base[8], di[8], dj[8];
    #pragma unroll
    for (int t = 0; t < 8; ++t) {
      int idx = leaf[c * 8 + t];
      int ch  = idx / 9;
      int p   = idx - ch * 9;
      int r   = p / 3;
      base[t] = ch * H * H;
      di[t]   = r - 1;
      dj[t]   = (p - r * 3) - 1;
    }
    float cf[16];
    #pragma unroll
    for (int t = 0; t < 16; ++t) cf[t] = coef[(size_t)c * 28 + t];
    float best = -3.0e38f;
    #pragma unroll
    for (int q = 0; q < 4; ++q) {
      const int i0 = 2 * pi + (q >> 1);
      const int j0 = 2 * pj + (q & 1);
      float v[8];
      #pragma unroll
      for (int t = 0; t < 8; ++t) {
        int ii = i0 + di[t], jj = j0 + dj[t];
        bool ok = (ii >= 0) & (ii < H) & (jj >= 0) & (jj < H);
        v[t] = ok ? s[base[t] + ii * H + jj] : 0.0f;
      }
      float g0 = gate4(cf[0],  cf[1],  cf[2],  cf[3],  v[0], v[1]);
      float g1 = gate4(cf[4],  cf[5],  cf[6],  cf[7],  v[2], v[3]);
      float g2 = gate4(cf[8],  cf[9],  cf[10], cf[11], v[4], v[5]);
      float g3 = gate4(cf[12], cf[13], cf[14], cf[15], v[6], v[7]);
      float h0 = gate4(cf[4],  cf[5],  cf[6],  cf[7],  g0, g1);
      float h1 = gate4(cf[8],  cf[9],  cf[10], cf[11], g2, g3);
      float r  = gate4(cf[12], cf[13], cf[14], cf[15], h0, h1);
      best = fmaxf(best, r);
    }
    if (transposed) out[(size_t)o * 128 + b] = best;   // [unit][batch]
    else            out[(size_t)b * total + o] = best; // [batch][c][pi][pj]
  }
}

// ---------------------------------------------------------------------------
// Stage 1: binarize x (3ch) -> 9ch {0,1} in LDS, conv-tree + pool.
// ---------------------------------------------------------------------------
__global__ void conv_stage1_kernel(const float* __restrict__ x,
                                   const int* __restrict__ leaf,
                                   const float* __restrict__ coef,
                                   float* __restrict__ out) {
  __shared__ float s[9 * 32 * 32];   // 36 KB
  const int b = blockIdx.x;
  const float* xb = x + (size_t)b * 3072;
  for (int i = threadIdx.x; i < 3072; i += blockDim.x) {
    float v = xb[i];
    int c = i >> 10, pix = i & 1023;
    s[(0 + c) * 1024 + pix] = (v > 0.25f) ? 1.0f : 0.0f;
    s[(3 + c) * 1024 + pix] = (v > 0.50f) ? 1.0f : 0.0f;
    s[(6 + c) * 1024 + pix] = (v > 0.75f) ? 1.0f : 0.0f;
  }
  __syncthreads();
  conv_eval(s, leaf, coef, out, b, 32, 32, 0);
}

// ---------------------------------------------------------------------------
// Stages 2-4: input plane is exactly 8192 floats (32KB) per image.
// Stage the plane into LDS with CDNA5 async global->LDS DMA (ASYNCcnt path).
// ---------------------------------------------------------------------------
__global__ void conv_stageN_kernel(const float* __restrict__ in,
                                   const int* __restrict__ leaf,
                                   const float* __restrict__ coef,
                                   float* __restrict__ out,
                                   int H, int OC, int transposed) {
  __shared__ float s[8192];          // 32 KB, == CIN*H*H for stages 2/3/4
  const int b = blockIdx.x;
  const unsigned gbase = (unsigned)b * 32768u;   // byte offset of image b
  for (int i = threadIdx.x; i < 2048; i += blockDim.x) {
    unsigned lds_addr = (unsigned)(unsigned long long)(&s[i * 4]); // LDS byte offset
    unsigned goff = gbase + (unsigned)i * 16u;
    asm volatile("global_load_async_to_lds_b128 %0, %1, %2"
                 :: "v"(lds_addr), "v"(goff), "s"(in) : "memory");
  }
  asm volatile("s_wait_asynccnt 0x0" ::: "memory");
  __syncthreads();
  conv_eval(s, leaf, coef, out, b, H, OC, transposed);
}

// ---------------------------------------------------------------------------
// FC difflogic layer on transposed activations [unit][batch].
// blockDim = 128 (one batch per thread) -> every gather is a coalesced line.
// ---------------------------------------------------------------------------
__global__ void fc_gate_kernel(const float* __restrict__ in_t,
                               const float* __restrict__ coef,
                               const int* __restrict__ ca,
                               const int* __restrict__ cb,
                               float* __restrict__ out_t) {
  const int u = blockIdx.x;
  const int b = threadIdx.x;
  const int ia = ca[u], ib = cb[u];
  const float c0 = coef[(size_t)u*4+0], c1 = coef[(size_t)u*4+1];
  const float c2 = coef[(size_t)u*4+2], c3 = coef[(size_t)u*4+3];
  const float xa = in_t[(size_t)ia * 128 + b];
  const float xb = in_t[(size_t)ib * 128 + b];
  out_t[(size_t)u * 128 + b] = c0 + c1 * xa + c2 * xb + c3 * (xa * xb);
}

// ---------------------------------------------------------------------------
// FC3 + class reduction with V_WMMA_F32_16X16X4_F32 (exact fp32 matmul).
// out(128x10) = G(128x10240) x S(10240x10), S block-diagonal ones, /TAU.
// grid (8, 10): blockIdx.x = 16-batch tile, blockIdx.y = class.
// A layout (16x4 f32): lane m=L&15; vgpr0=K(2*hi), vgpr1=K(2*hi+1), hi=L>>4.
// B rows are identical (col cls = 1), so per-lane B value = (N==cls),
// robust to B's K-row striping. C/D: vgpr v, M=v+8*hi, N=L&15.
// ---------------------------------------------------------------------------
typedef float v2f __attribute__((ext_vector_type(2)));
typedef float v8f __attribute__((ext_vector_type(8)));

__global__ void fc3_reduce_wmma_kernel(const float* __restrict__ h2t,
                                       const float* __restrict__ coef,
                                       const int* __restrict__ ca,
                                       const int* __restrict__ cb,
                                       float* __restrict__ out) {
  const int lane = threadIdx.x;       // 0..31, one wave per block
  const int m  = lane & 15;
  const int hi = lane >> 4;
  const int b0 = blockIdx.x * 16;
  const int cls = blockIdx.y;         // 0..9
  const int bb = b0 + m;
  v2f bf;
  bf.x = (m == cls) ? 1.0f : 0.0f;
  bf.y = bf.x;
  v8f acc = {0.f, 0.f, 0.f, 0.f, 0.f, 0.f, 0.f, 0.f};
  const int kbeg = cls * 1024;
  for (int k = kbeg; k < kbeg + 1024; k += 4) {
    const int u0 = k + 2 * hi;
    const int u1 = u0 + 1;
    const float xa0 = h2t[(size_t)ca[u0] * 128 + bb];
    const float xb0 = h2t[(size_t)cb[u0] * 128 + bb];
    const float xa1 = h2t[(size_t)ca[u1] * 128 + bb];
    const float xb1 = h2t[(size_t)cb[u1] * 128 + bb];
    const float* cp0 = coef + (size_t)u0 * 4;
    const float* cp1 = coef + (size_t)u1 * 4;
    v2f a;
    a.x = cp0[0] + cp0[1] * xa0 + cp0[2] * xb0 + cp0[3] * (xa0 * xb0);
    a.y = cp1[0] + cp1[1] * xa1 + cp1[2] * xb1 + cp1[3] * (xa1 * xb1);
    acc = __builtin_amdgcn_wmma_f32_16x16x4_f32(false, a, false, bf,
                                                (short)0, acc, false, false);
  }
  if (m == cls) {                      // lanes holding column N == cls
    #pragma unroll
    for (int v = 0; v < 8; ++v) {
      out[(size_t)(b0 + v + hi * 8) * 10 + cls] = acc[v] / 10.0f;  // /TAU
    }
  }
}

// ---------------------------------------------------------------------------
// Host-side orchestration
// ---------------------------------------------------------------------------
extern "C" void kernel_launch(void* const* d_in, const int* in_sizes, int n_in,
                              void* d_out, int out_size, void* d_ws, size_t ws_size,
                              hipStream_t stream) {
  const float* x   = (const float*)d_in[0];
  const float* w1  = (const float*)d_in[1];
  const float* w2  = (const float*)d_in[2];
  const float* w3  = (const float*)d_in[3];
  const float* w4  = (const float*)d_in[4];
  const float* fw1 = (const float*)d_in[5];
  const float* fw2 = (const float*)d_in[6];
  const float* fw3 = (const float*)d_in[7];
  const int* l1  = (const int*)d_in[8];
  const int* l2  = (const int*)d_in[9];
  const int* l3  = (const int*)d_in[10];
  const int* l4  = (const int*)d_in[11];
  const int* ca1 = (const int*)d_in[12];
  const int* cb1 = (const int*)d_in[13];
  const int* ca2 = (const int*)d_in[14];
  const int* cb2 = (const int*)d_in[15];
  const int* ca3 = (const int*)d_in[16];
  const int* cb3 = (const int*)d_in[17];
  float* out = (float*)d_out;
  float* ws  = (float*)d_ws;

  // Workspace partition (floats); all regions fully overwritten per call.
  float* c1  = ws;                 // 32*7*4    = 896
  float* c2  = c1  + 896;          // 128*7*4   = 3584
  float* c3  = c2  + 3584;         // 512*7*4   = 14336
  float* c4  = c3  + 14336;        // 1024*7*4  = 28672
  float* cf1 = c4  + 28672;        // 40960*4   = 163840
  float* cf2 = cf1 + 163840;       // 20480*4   = 81920
  float* cf3 = cf2 + 81920;        // 10240*4   = 40960
  float* a1  = cf3 + 40960;        // 128*32*16*16 = 1048576   [b][c][h][w]
  float* a2  = a1  + 1048576;      // 128*128*8*8  = 1048576
  float* a3  = a2  + 1048576;      // 128*512*4*4  = 1048576
  float* a4t = a3  + 1048576;      // 4096*128     = 524288    [unit][b]
  float* h1t = a4t + 524288;       // 40960*128    = 5242880   [unit][b]
  float* h2t = h1t + 5242880;      // 20480*128    = 2621440   [unit][b]
  // total ~45.3 MB

  // 1) Gate coefficients: softmax(w) @ COEF
  softmax_coef_kernel<<<(224   + 255)/256, 256, 0, stream>>>(w1,  c1,  224);
  softmax_coef_kernel<<<(896   + 255)/256, 256, 0, stream>>>(w2,  c2,  896);
  softmax_coef_kernel<<<(3584  + 255)/256, 256, 0, stream>>>(w3,  c3,  3584);
  softmax_coef_kernel<<<(7168  + 255)/256, 256, 0, stream>>>(w4,  c4,  7168);
  softmax_coef_kernel<<<(40960 + 255)/256, 256, 0, stream>>>(fw1, cf1, 40960);
  softmax_coef_kernel<<<(20480 + 255)/256, 256, 0, stream>>>(fw2, cf2, 20480);
  softmax_coef_kernel<<<(10240 + 255)/256, 256, 0, stream>>>(fw3, cf3, 10240);

  // 2) Conv-tree stages (one image per blockIdx.x, OC split over blockIdx.y)
  conv_stage1_kernel<<<dim3(128, 4), 256, 0, stream>>>(x, l1, c1, a1);
  conv_stageN_kernel<<<dim3(128, 4), 256, 0, stream>>>(a1, l2, c2, a2,  16,  128, 0);
  conv_stageN_kernel<<<dim3(128, 4), 256, 0, stream>>>(a2, l3, c3, a3,   8,  512, 0);
  conv_stageN_kernel<<<dim3(128, 4), 256, 0, stream>>>(a3, l4, c4, a4t,  4, 1024, 1);

  // 3) FC difflogic layers on transposed activations
  fc_gate_kernel<<<40960, 128, 0, stream>>>(a4t, cf1, ca1, cb1, h1t);
  fc_gate_kernel<<<20480, 128, 0, stream>>>(h1t, cf2, ca2, cb2, h2t);

  // 4) FC3 fused with class-sum via fp32 WMMA, /TAU
  fc3_reduce_wmma_kernel<<<dim3(8, 10), 32, 0, stream>>>(h2t, cf3, ca3, cb3, out);
}